// GCNGPP_13683765805700
// MI455X (gfx1250) — compile-verified
//
#include <hip/hip_runtime.h>

#define NN 100000      // nodes
#define NE 1600000     // edges
#define NG 128         // graphs
#define D  128         // embedding dim
#define NCLS 8
#define BN_EPS 1e-5f

typedef float v2f __attribute__((ext_vector_type(2)));
typedef float v8f __attribute__((ext_vector_type(8)));

// ---------- setup kernels ----------
__global__ void k_init_deg(float* deg) {
    int n = blockIdx.x * 256 + threadIdx.x;
    if (n < NN) deg[n] = 1.0f;   // degree(row)+1
}

__global__ void k_deg_accum(const int* __restrict__ ei, float* deg) {
    int e = blockIdx.x * 256 + threadIdx.x;
    if (e < NE) atomicAdd(&deg[ei[e]], 1.0f);   // row = ei[0][e]
}

__global__ void k_embed(const int* __restrict__ x, const float* __restrict__ table,
                        float* __restrict__ h) {
    int i = blockIdx.x * 256 + threadIdx.x;     // over NN*D
    int n = i >> 7, c = i & 127;
    if (n < NN) h[i] = table[x[n] * D + c];
}

__global__ void k_dinv(const float* __restrict__ deg, float* __restrict__ dinv) {
    int n = blockIdx.x * 256 + threadIdx.x;
    if (n < NN) dinv[n] = rsqrtf(deg[n]);
}

__global__ void k_zero(float* p, int n) {
    int i = blockIdx.x * 256 + threadIdx.x;
    if (i < n) p[i] = 0.0f;
}

// ---------- node GEMM: hl = h @ W + b, via V_WMMA_F32_16X16X4_F32 ----------
// grid = NN/16 blocks of 256 threads (8 waves). Wave w computes the 16x16 tile
// at rows m0=16*blockIdx, cols n0=16*w, looping K=128 in steps of 4.
__global__ void k_gemm_wmma(const float* __restrict__ h, const float* __restrict__ W,
                            const float* __restrict__ bias, float* __restrict__ hl) {
    const int lane = threadIdx.x & 31;
    const int wave = threadIdx.x >> 5;
    const int m0 = blockIdx.x * 16;
    const int n0 = wave * 16;
    const int l15 = lane & 15;
    const int hi  = lane >> 4;            // 0: lanes 0-15, 1: lanes 16-31

    // A fragment: lane holds h[m0+l15][k + 2*hi .. +1]  (16x4 f32 layout)
    const float* aptr = h + (size_t)(m0 + l15) * D + (hi << 1);
    // B fragment: v.x = W[k + 2*hi][n0+l15], v.y = W[k + 2*hi + 1][n0+l15]
    const float* bptr = W + (size_t)(hi << 1) * D + n0 + l15;

    v8f acc = {};
    #pragma unroll
    for (int k = 0; k < D; k += 4) {
        v2f a = *(const v2f*)(aptr + k);
        v2f b;
        b.x = bptr[(size_t)k * D];
        b.y = bptr[(size_t)(k + 1) * D];
        acc = __builtin_amdgcn_wmma_f32_16x16x4_f32(
            false, a, false, b, (short)0, acc, false, false);
    }

    // C/D layout: VGPR r -> row m0+r (lanes 0-15) / m0+r+8 (lanes 16-31), col n0+l15
    float bv = bias[n0 + l15];
    float* op = hl + (size_t)(m0 + (hi << 3)) * D + n0 + l15;
    #pragma unroll
    for (int r = 0; r < 8; r++) op[(size_t)r * D] = acc[r] + bv;
}

// ---------- edge message + scatter: aggr[col] += norm * relu(hl[row] + ee) ----------
__global__ void k_edge(const int* __restrict__ ei, const float* __restrict__ eattr,
                       const float* __restrict__ dinv, const float* __restrict__ hl,
                       const float* __restrict__ eW, const float* __restrict__ eb,
                       float* __restrict__ aggr) {
    long long idx = (long long)blockIdx.x * 256 + threadIdx.x;   // NE * D threads
    if (idx >= (long long)NE * D) return;
    int e = (int)(idx >> 7);
    int c = (int)(idx & 127);
    int row = ei[e];
    int col = ei[NE + e];
    float norm = dinv[row] * dinv[col];
    float4 ea = *(const float4*)(eattr + (size_t)e * 4);
    float ee = eb[c] + ea.x * eW[c] + ea.y * eW[D + c]
                     + ea.z * eW[2 * D + c] + ea.w * eW[3 * D + c];
    float v = hl[(size_t)row * D + c] + ee;
    v = fmaxf(v, 0.0f);
    atomicAdd(&aggr[(size_t)col * D + c], norm * v);
}

// ---------- combine self-loop term (in-place on aggr) + BN stat accumulation ----------
// block handles 128 nodes x 128 channels; thread t: channel c=t&127, node parity r=t>>7
__global__ void k_combine(float* __restrict__ aggr, const float* __restrict__ hl,
                          const float* __restrict__ root, const float* __restrict__ deg,
                          float* __restrict__ stats) {
    __shared__ float s1[256];
    __shared__ float s2[256];
    int c = threadIdx.x & 127;
    int r = threadIdx.x >> 7;
    int base = blockIdx.x * 128;
    float sum = 0.0f, ssq = 0.0f;
    float rc = root[c];
    for (int i = 0; i < 64; i++) {
        int n = base + (i << 1) + r;
        if (n < NN) {
            size_t off = (size_t)n * D + c;
            float self = fmaxf(hl[off] + rc, 0.0f) / deg[n];
            float o = aggr[off] + self;
            aggr[off] = o;
            sum += o;
            ssq += o * o;
        }
    }
    s1[threadIdx.x] = sum;
    s2[threadIdx.x] = ssq;
    __syncthreads();
    if (r == 0) {
        atomicAdd(&stats[c],     s1[c] + s1[c + 128]);
        atomicAdd(&stats[D + c], s2[c] + s2[c + 128]);
    }
}

__global__ void k_bn_finalize(const float* __restrict__ gamma, const float* __restrict__ beta,
                              float* __restrict__ stats) {
    int c = threadIdx.x;   // 128 threads
    float mean = stats[c] / (float)NN;
    float var  = stats[D + c] / (float)NN - mean * mean;
    float rstd = rsqrtf(var + BN_EPS);
    float sc = rstd * gamma[c];
    stats[2 * D + c] = sc;
    stats[3 * D + c] = beta[c] - mean * sc;
}

__global__ void k_bn_apply(float* __restrict__ buf, const float* __restrict__ stats,
                           int do_relu) {
    int i = blockIdx.x * 256 + threadIdx.x;
    if (i >= NN * D) return;
    int c = i & 127;
    float v = buf[i] * stats[2 * D + c] + stats[3 * D + c];
    if (do_relu) v = fmaxf(v, 0.0f);
    buf[i] = v;
}

// ---------- global mean pool ----------
__global__ void k_pool(const float* __restrict__ h, const int* __restrict__ batch,
                       float* __restrict__ pool, float* __restrict__ cnt) {
    int i = blockIdx.x * 256 + threadIdx.x;
    if (i >= NN * D) return;
    int n = i >> 7, c = i & 127;
    int g = batch[n];
    atomicAdd(&pool[(size_t)g * D + c], h[i]);
    if (c == 0) atomicAdd(&cnt[g], 1.0f);
}

// ---------- output head: out = (pool/cnt) @ out_W + out_b ----------
__global__ void k_head(const float* __restrict__ pool, const float* __restrict__ cnt,
                       const float* __restrict__ oW, const float* __restrict__ ob,
                       float* __restrict__ out) {
    int i = blockIdx.x * 256 + threadIdx.x;
    if (i >= NG * NCLS) return;
    int g = i / NCLS, j = i % NCLS;
    float inv = 1.0f / fmaxf(cnt[g], 1.0f);
    float acc = ob[j];
    for (int c = 0; c < D; c++)
        acc += pool[(size_t)g * D + c] * inv * oW[c * NCLS + j];
    out[i] = acc;
}

extern "C" void kernel_launch(void* const* d_in, const int* in_sizes, int n_in,
                              void* d_out, int out_size, void* d_ws, size_t ws_size,
                              hipStream_t stream) {
    const int*   x        = (const int*)  d_in[0];
    const int*   ei       = (const int*)  d_in[1];   // [2, NE]
    const float* eattr    = (const float*)d_in[2];   // [NE, 4]
    const int*   batch    = (const int*)  d_in[3];
    const float* table    = (const float*)d_in[4];   // [128, D]
    const float* lin_W    = (const float*)d_in[5];   // [2, D, D]
    const float* lin_b    = (const float*)d_in[6];   // [2, D]
    const float* root_emb = (const float*)d_in[7];   // [2, D]
    const float* edge_W   = (const float*)d_in[8];   // [2, 4, D]
    const float* edge_b   = (const float*)d_in[9];   // [2, D]
    const float* bn_gamma = (const float*)d_in[10];  // [2, D]
    const float* bn_beta  = (const float*)d_in[11];  // [2, D]
    const float* out_W    = (const float*)d_in[12];  // [D, NCLS]
    const float* out_b    = (const float*)d_in[13];  // [NCLS]
    float* out = (float*)d_out;

    // workspace carve-up (256B aligned slices)
    char* ws = (char*)d_ws;
    size_t off = 0;
    auto carve = [&](size_t bytes) -> float* {
        float* p = (float*)(ws + off);
        off += (bytes + 255) & ~(size_t)255;
        return p;
    };
    float* bufH  = carve((size_t)NN * D * 4);   // h / aggr / out (reused)
    float* bufHL = carve((size_t)NN * D * 4);   // hl
    float* deg   = carve((size_t)NN * 4);
    float* dinv  = carve((size_t)NN * 4);
    float* stats = carve(4 * D * 4);            // sum | sumsq | scale | shift
    float* pool  = carve((size_t)NG * D * 4);
    float* cnt   = carve((size_t)NG * 4);
    (void)ws_size; (void)n_in; (void)in_sizes; (void)out_size;

    const int NB_N   = (NN + 255) / 256;        // node-count grids
    const int NB_E   = (NE + 255) / 256;
    const int NB_ND  = (NN * D) / 256;          // 50000 (exact)
    const long long edge_threads = (long long)NE * D;
    const int NB_ED  = (int)((edge_threads + 255) / 256);   // 800000

    // degree + embedding + inverse-sqrt
    k_init_deg <<<NB_N, 256, 0, stream>>>(deg);
    k_deg_accum<<<NB_E, 256, 0, stream>>>(ei, deg);
    k_embed    <<<NB_ND, 256, 0, stream>>>(x, table, bufH);
    k_dinv     <<<NB_N, 256, 0, stream>>>(deg, dinv);

    for (int l = 0; l < 2; l++) {
        // hl = h @ lin_W[l] + lin_b[l]   (WMMA f32)
        k_gemm_wmma<<<NN / 16, 256, 0, stream>>>(bufH, lin_W + (size_t)l * D * D,
                                                 lin_b + l * D, bufHL);
        // h consumed -> reuse bufH as scatter accumulator
        k_zero<<<NB_ND, 256, 0, stream>>>(bufH, NN * D);
        k_zero<<<2, 256, 0, stream>>>(stats, 2 * D);
        // edge messages + scatter-add
        k_edge<<<NB_ED, 256, 0, stream>>>(ei, eattr, dinv, bufHL,
                                          edge_W + (size_t)l * 4 * D,
                                          edge_b + l * D, bufH);
        // + relu(hl + root)/deg, accumulate BN stats
        k_combine<<<(NN + 127) / 128, 256, 0, stream>>>(bufH, bufHL,
                                                        root_emb + l * D, deg, stats);
        k_bn_finalize<<<1, 128, 0, stream>>>(bn_gamma + l * D, bn_beta + l * D, stats);
        k_bn_apply<<<NB_ND, 256, 0, stream>>>(bufH, stats, (l == 0) ? 1 : 0);
    }

    // global mean pool + output head
    k_zero<<<(NG * D + NG + 255) / 256, 256, 0, stream>>>(pool, NG * D + NG); // pool & cnt adjacent: cnt = pool + NG*D
    k_zero<<<1, 256, 0, stream>>>(cnt, NG);
    k_pool<<<NB_ND, 256, 0, stream>>>(bufH, batch, pool, cnt);
    k_head<<<(NG * NCLS + 255) / 256, 256, 0, stream>>>(pool, cnt, out_W, out_b, out);
}